// DGCNN_Embedder_88759794139828
// MI455X (gfx1250) — compile-verified
//
#include <hip/hip_runtime.h>
#include <hip/hip_bf16.h>
#include <cstdint>

typedef __attribute__((ext_vector_type(2))) float v2f;
typedef __attribute__((ext_vector_type(8))) float v8f;

#define NPTS  2048
#define BATCH 4
#define KNN   20
#define SLOPE 0.2f
#define EPSV  1e-5f
#define NEGBIG -3.0e38f

// CDNA5 f32 WMMA: D(16x16) = A(16x4) * B(4x16) + C, wave32.
// A/B frag: lane l holds (row = l%16, K = kt + 2*(l/16) + {0,1})  -> one float2
// C/D frag: lane l, vgpr v holds (M = v + 8*(l/16), N = l%16)
__device__ __forceinline__ v8f wmma4f32(v2f a, v2f b, v8f c) {
  return __builtin_amdgcn_wmma_f32_16x16x4_f32(false, a, false, b, (short)0, c,
                                               false, false);
}

// gfx1250 async DMA: global -> LDS, 16B per lane per issue; tracked by ASYNCcnt
__device__ __forceinline__ void async_g2lds_b128(uint32_t lds_off,
                                                 const void* gaddr) {
  asm volatile("global_load_async_to_lds_b128 %0, %1, off" ::"v"(lds_off),
               "v"(gaddr)
               : "memory");
}
__device__ __forceinline__ void wait_asynccnt0() {
  asm volatile("s_wait_asynccnt 0x0" ::: "memory");
}

// ---- Build stage-1 features: (B,3,N) -> point-major (B*N, 4) zero padded ----
__global__ void k_build_x0(const float* __restrict__ x, float* __restrict__ X0) {
  int tid = blockIdx.x * blockDim.x + threadIdx.x;  // b*N + n
  int b = tid / NPTS, n = tid % NPTS;
  float4 v;
  v.x = x[(b * 3 + 0) * NPTS + n];
  v.y = x[(b * 3 + 1) * NPTS + n];
  v.z = x[(b * 3 + 2) * NPTS + n];
  v.w = 0.f;
  *(float4*)(X0 + (size_t)tid * 4) = v;
}

// ---- Per-point squared norm ----
__global__ void k_rowsq(const float* __restrict__ X, int ldx, int Kc,
                        float* __restrict__ xx) {
  int tid = blockIdx.x * blockDim.x + threadIdx.x;  // b*N + n
  const float* row = X + (size_t)tid * ldx;
  float s = 0.f;
  for (int c = 0; c < Kc; ++c) { float v = row[c]; s += v * v; }
  xx[tid] = s;
}

// ---- Fused: distance strip (16 rows x 2048) in LDS via WMMA, then top-K ----
// One block owns rows nt*16..nt*16+15 of batch b. 144 KB LDS / block.
__global__ void __launch_bounds__(256)
k_dist_topk(const float* __restrict__ X, int ldx, int Kc,
            const float* __restrict__ xx, int* __restrict__ idx) {
  __shared__ float sd[16 * NPTS];  // 128 KB: 16 distance rows
  __shared__ float sAx[16 * 256];  // 16 KB: 16 A-side feature rows (Kc<=256)
  const int tid = threadIdx.x;
  const int lane = tid & 31, w = tid >> 5;
  const int half = lane >> 4, r = lane & 15;
  const int nt = blockIdx.x, b = blockIdx.y;
  const float* Xb = X + (size_t)b * NPTS * ldx;

  // Async-DMA the 16 A-side rows into LDS (shared by all 8 waves, all 128 mt)
  {
    uint32_t l0 = (uint32_t)(uintptr_t)sAx;
    const int cpr = Kc / 4;  // 16B chunks per row
    for (int c = tid; c < 16 * cpr; c += 256) {
      int row = c / cpr, off = c % cpr;
      uint32_t lo = l0 + (uint32_t)(row * Kc + off * 4) * 4u;
      const float* gp = Xb + (size_t)(nt * 16 + row) * ldx + off * 4;
      async_g2lds_b128(lo, gp);
    }
    wait_asynccnt0();
  }
  __syncthreads();

  const float* xxb = xx + b * NPTS;
  float xr8[8];
#pragma unroll
  for (int v = 0; v < 8; ++v) xr8[v] = xxb[nt * 16 + 8 * half + v];
  const float* arow = sAx + r * Kc + 2 * half;

  for (int mt = w; mt < NPTS / 16; mt += 8) {  // uniform per wave
    const float* brow = Xb + (size_t)(mt * 16 + r) * ldx + 2 * half;
    v8f acc = {0.f, 0.f, 0.f, 0.f, 0.f, 0.f, 0.f, 0.f};
    int kt = 0;
    for (; kt + 32 <= Kc; kt += 32) {  // batch 16 loads, then 8 WMMAs
      v2f av[8], bv[8];
#pragma unroll
      for (int i = 0; i < 8; ++i) {
        av[i] = *(const v2f*)(arow + kt + 4 * i);
        bv[i] = *(const v2f*)(brow + kt + 4 * i);
      }
#pragma unroll
      for (int i = 0; i < 8; ++i) acc = wmma4f32(av[i], bv[i], acc);
    }
    for (; kt < Kc; kt += 4) {
      v2f av = *(const v2f*)(arow + kt);
      v2f bv = *(const v2f*)(brow + kt);
      acc = wmma4f32(av, bv, acc);
    }
    const int col = mt * 16 + r;
    const float xc = xxb[col];
#pragma unroll
    for (int v = 0; v < 8; ++v)
      sd[(v + 8 * half) * NPTS + col] = 2.f * acc[v] - xr8[v] - xc;
  }
  __syncthreads();

  // Top-K straight out of LDS: wave w handles rows w and w+8
  for (int rr0 = 0; rr0 < 2; ++rr0) {
    const int rr = w + rr0 * 8;
    float* s = sd + rr * NPTS;
    int* out = idx + ((size_t)b * NPTS + nt * 16 + rr) * KNN;
    for (int t = 0; t < KNN; ++t) {
      float v = NEGBIG;
      int bi = 0;
      for (int j = 0; j < NPTS / 32; ++j) {
        int id = lane + j * 32;
        float xv = s[id];
        if (xv > v) { v = xv; bi = id; }  // strict > : smallest id per lane
      }
      for (int off = 16; off; off >>= 1) {  // wave32 butterfly argmax
        float ov = __shfl_xor(v, off, 32);
        int oi = __shfl_xor(bi, off, 32);
        if (ov > v || (ov == v && oi < bi)) { v = ov; bi = oi; }
      }
      if (lane == 0) { out[t] = bi; s[bi] = NEGBIG; }
      __syncthreads();  // uniform: every thread hits 2*KNN barriers
    }
  }
}

// ---- Split W(O,2C) into Wa = W[:, :C], Wd = W[:, C:] - W[:, :C] (padded) ----
__global__ void k_prep_w(const float* __restrict__ W, float* __restrict__ Wa,
                         float* __restrict__ Wd, int O, int Cin, int Cpad) {
  int t = blockIdx.x * blockDim.x + threadIdx.x;
  if (t >= O * Cpad) return;
  int o = t / Cpad, c = t % Cpad;
  float a = 0.f, bb = 0.f;
  if (c < Cin) { a = W[o * 2 * Cin + c]; bb = W[o * 2 * Cin + Cin + c]; }
  Wa[t] = a;
  Wd[t] = bb - a;
}

// ---- Dense GEMM out(b,n,o) = A(O,Kc,row-major,ld==Kc) x X(b,n,Kc) ----
// Weight tile (16 x Kc) async-DMA'd to LDS once per block (8 waves share it).
__global__ void __launch_bounds__(256)
k_gemm(const float* __restrict__ A, int Kc, const float* __restrict__ X,
       int ldx, float* __restrict__ out, int O) {
  __shared__ float sA[16 * 512];  // 32 KB max (Kc <= 512)
  const int tid = threadIdx.x;
  const int lane = tid & 31, w = tid >> 5;
  const int half = lane >> 4, r = lane & 15;
  const int ntile = blockIdx.x * 8 + w;
  const int ot = blockIdx.y, b = blockIdx.z;
  {
    const char* Ab = (const char*)(A + (size_t)ot * 16 * Kc);  // contiguous
    uint32_t l0 = (uint32_t)(uintptr_t)sA;
    const int nch = 4 * Kc;  // 16B chunks in the 16*Kc tile
    for (int c = tid; c < nch; c += 256)
      async_g2lds_b128(l0 + (uint32_t)c * 16u, Ab + (size_t)c * 16);
    wait_asynccnt0();
  }
  __syncthreads();

  const float* arow = sA + r * Kc + 2 * half;
  const float* brow = X + ((size_t)b * NPTS + ntile * 16 + r) * ldx + 2 * half;
  v8f acc = {0.f, 0.f, 0.f, 0.f, 0.f, 0.f, 0.f, 0.f};
  int kt = 0;
  for (; kt + 32 <= Kc; kt += 32) {
    v2f av[8], bv[8];
#pragma unroll
    for (int i = 0; i < 8; ++i) {
      av[i] = *(const v2f*)(arow + kt + 4 * i);
      bv[i] = *(const v2f*)(brow + kt + 4 * i);
    }
#pragma unroll
    for (int i = 0; i < 8; ++i) acc = wmma4f32(av[i], bv[i], acc);
  }
  for (; kt < Kc; kt += 4) {
    v2f av = *(const v2f*)(arow + kt);
    v2f bv = *(const v2f*)(brow + kt);
    acc = wmma4f32(av, bv, acc);
  }
  int n = ntile * 16 + r;
  int obase = ot * 16 + 8 * half;
  float* dst = out + ((size_t)b * NPTS + n) * O + obase;
  *(float4*)(dst) = make_float4(acc[0], acc[1], acc[2], acc[3]);
  *(float4*)(dst + 4) = make_float4(acc[4], acc[5], acc[6], acc[7]);
}

// ---- Gather pass: BN stats over (b,n,k) and hmax = G2 + max_k G1[idx] ----
__global__ void k_edge_stats(const float* __restrict__ G1,
                             const float* __restrict__ G2,
                             const int* __restrict__ idx, int O,
                             float* __restrict__ hm, float* __restrict__ sum,
                             float* __restrict__ sumsq) {
  const int o = threadIdx.x;         // blockDim.x == O
  const int base = blockIdx.x * 16;  // chunk of 16 rows, same batch
  const int b = base / NPTS;
  float accS = 0.f, accQ = 0.f;
  for (int i = 0; i < 16; ++i) {
    int rown = base + i;  // b*N + n
    float g2 = G2[(size_t)rown * O + o];
    const int* id = idx + (size_t)rown * KNN;
    float mx = NEGBIG;
    for (int k = 0; k < KNN; ++k) {
      int j = id[k];
      float v = G1[((size_t)b * NPTS + j) * O + o];
      mx = fmaxf(mx, v);
      float h = v + g2;
      accS += h;
      accQ += h * h;
    }
    hm[(size_t)rown * O + o] = mx + g2;
  }
  atomicAdd(&sum[o], accS);
  atomicAdd(&sumsq[o], accQ);
}

// ---- BN + LReLU (monotone => commutes with max_k), write into cat slice ----
__global__ void k_edge_final(const float* __restrict__ hm,
                             const float* __restrict__ sum,
                             const float* __restrict__ sumsq,
                             const float* __restrict__ g,
                             const float* __restrict__ be, int O, int coff,
                             float* __restrict__ cat) {
  int t = blockIdx.x * blockDim.x + threadIdx.x;  // b*N*O
  int o = t % O;
  int rown = t / O;
  const float cnt = (float)(BATCH * NPTS * KNN);
  float m = sum[o] / cnt;
  float var = sumsq[o] / cnt - m * m;
  float sc = rsqrtf(var + EPSV) * g[o];
  float y = (hm[t] - m) * sc + be[o];
  y = y >= 0.f ? y : SLOPE * y;
  cat[(size_t)rown * 512 + coff + o] = y;
}

// ---- BN stats for final pointwise layer (O=1024, count = B*N) ----
__global__ void k_pf_stats(const float* __restrict__ pf,
                           float* __restrict__ sum, float* __restrict__ sumsq) {
  int o = blockIdx.y * 256 + threadIdx.x;
  int base = blockIdx.x * 8;  // 8 rows (b*N+n)
  float s = 0.f, q = 0.f;
  for (int i = 0; i < 8; ++i) {
    float v = pf[(size_t)(base + i) * 1024 + o];
    s += v;
    q += v * v;
  }
  atomicAdd(&sum[o], s);
  atomicAdd(&sumsq[o], q);
}

// ---- Final: lrelu(bn(pf)); out0 = pf[...,nb=0,:]; group = concat([m,m]) ----
__global__ void __launch_bounds__(128)
k_pf_final(const float* __restrict__ pf, const float* __restrict__ sum,
           const float* __restrict__ sumsq, const float* __restrict__ g,
           const float* __restrict__ be, float* __restrict__ out) {
  __shared__ float red[128];
  int p = threadIdx.x, nb = blockIdx.x, o = blockIdx.y, b = blockIdx.z;
  int n = nb * 128 + p;
  const float cnt = (float)(BATCH * NPTS);
  float m = sum[o] / cnt;
  float var = sumsq[o] / cnt - m * m;
  float sc = rsqrtf(var + EPSV) * g[o];
  float v = pf[((size_t)b * NPTS + n) * 1024 + o];
  float y = (v - m) * sc + be[o];
  y = y >= 0.f ? y : SLOPE * y;
  if (nb == 0) out[((size_t)b * 1024 + o) * 128 + p] = y;
  red[p] = y;
  __syncthreads();
  for (int s2 = 64; s2; s2 >>= 1) {
    if (p < s2) red[p] = fmaxf(red[p], red[p + s2]);
    __syncthreads();
  }
  if (p == 0) {
    float mx = red[0];
    float* grp = out + (size_t)BATCH * 1024 * 128;
    grp[((size_t)b * 2048 + o) * 16 + nb] = mx;
    grp[((size_t)b * 2048 + 1024 + o) * 16 + nb] = mx;
  }
}

extern "C" void kernel_launch(void* const* d_in, const int* in_sizes, int n_in,
                              void* d_out, int out_size, void* d_ws,
                              size_t ws_size, hipStream_t stream) {
  (void)in_sizes; (void)n_in; (void)out_size; (void)ws_size;
  const float* x = (const float*)d_in[0];
  const float* W[6] = {nullptr,
                       (const float*)d_in[1], (const float*)d_in[2],
                       (const float*)d_in[3], (const float*)d_in[4],
                       (const float*)d_in[5]};
  const float* gp[5] = {(const float*)d_in[6],  (const float*)d_in[8],
                        (const float*)d_in[10], (const float*)d_in[12],
                        (const float*)d_in[14]};
  const float* bp[5] = {(const float*)d_in[7],  (const float*)d_in[9],
                        (const float*)d_in[11], (const float*)d_in[13],
                        (const float*)d_in[15]};
  char* ws = (char*)d_ws;
  float* pf   = (float*)(ws + 0);         // 32 MB (B,N,1024)
  float* cat  = (float*)(ws + 33554432);  // 16 MB (B,N,512)
  float* G1   = (float*)(ws + 50331648);  // 8 MB
  float* G2   = (float*)(ws + 58720256);  // 8 MB
  float* hm   = (float*)(ws + 67108864);  // 8 MB
  float* X0   = (float*)(ws + 75497472);  // 128 KB
  float* xx   = (float*)(ws + 75628544);  // 32 KB
  int*   idxb = (int*)  (ws + 75661312);  // 640 KB
  float* Wa   = (float*)(ws + 76316672);  // 256 KB
  float* Wd   = (float*)(ws + 76578816);  // 256 KB
  float* sumA = (float*)(ws + 76840960);  // 4 KB
  float* sumB = (float*)(ws + 76845056);  // 4 KB

  k_build_x0<<<BATCH * NPTS / 256, 256, 0, stream>>>(x, X0);

  struct St { const float* Xin; int ldx, Cin, Cpad, O, coff, wi; };
  St st[4] = {
      {X0,        4,   3,   4,   64,  0,   1},
      {cat + 0,   512, 64,  64,  64,  64,  2},
      {cat + 64,  512, 64,  64,  128, 128, 3},
      {cat + 128, 512, 128, 128, 256, 256, 4},
  };
  for (int s = 0; s < 4; ++s) {
    const St& S = st[s];
    k_rowsq<<<BATCH * NPTS / 256, 256, 0, stream>>>(S.Xin, S.ldx, S.Cpad, xx);
    k_dist_topk<<<dim3(NPTS / 16, BATCH), 256, 0, stream>>>(S.Xin, S.ldx,
                                                            S.Cpad, xx, idxb);
    k_prep_w<<<(S.O * S.Cpad + 255) / 256, 256, 0, stream>>>(
        W[S.wi], Wa, Wd, S.O, S.Cin, S.Cpad);
    k_gemm<<<dim3(16, S.O / 16, BATCH), 256, 0, stream>>>(Wa, S.Cpad, S.Xin,
                                                          S.ldx, G1, S.O);
    k_gemm<<<dim3(16, S.O / 16, BATCH), 256, 0, stream>>>(Wd, S.Cpad, S.Xin,
                                                          S.ldx, G2, S.O);
    hipMemsetAsync(sumA, 0, 4096, stream);
    hipMemsetAsync(sumB, 0, 4096, stream);
    k_edge_stats<<<BATCH * NPTS / 16, S.O, 0, stream>>>(G1, G2, idxb, S.O, hm,
                                                        sumA, sumB);
    k_edge_final<<<BATCH * NPTS * S.O / 256, 256, 0, stream>>>(
        hm, sumA, sumB, gp[s], bp[s], S.O, S.coff, cat);
  }

  // Final pointwise layer: pf = W5(1024,512) x cat
  k_gemm<<<dim3(16, 64, BATCH), 256, 0, stream>>>(W[5], 512, cat, 512, pf,
                                                  1024);
  hipMemsetAsync(sumA, 0, 4096, stream);
  hipMemsetAsync(sumB, 0, 4096, stream);
  k_pf_stats<<<dim3(BATCH * NPTS / 8, 4), 256, 0, stream>>>(pf, sumA, sumB);
  k_pf_final<<<dim3(16, 1024, BATCH), 128, 0, stream>>>(
      pf, sumA, sumB, gp[4], bp[4], (float*)d_out);
}